// DePatchEmbed_80298708566630
// MI455X (gfx1250) — compile-verified
//
#include <hip/hip_runtime.h>
#include <stdint.h>

// DePatchEmbed: out[n, c, 2i+pi, 2j+pj] = x[n, j*128+i, 4c+2pi+pj]
//   x:   [32, 16384, 256] f32   (patch index k = j*128 + i, column-major patches)
//   out: [32, 64, 256, 256] f32
// Pure permutation, 1 GiB total traffic -> HBM-bound (~46us @ 23.3 TB/s).
// Strategy: per-workgroup tile (n, i, 32 j's) staged through LDS in OUTPUT
// layout using the CDNA5 async DMA path:
//   global_load_async_to_lds_b64   (8B chunks: the pj-pair is contiguous on
//                                   both the input and output side)
//   s_wait_asynccnt 0 ; barrier
//   global_store_async_from_lds_b128 (256B contiguous output row segments)
// No VALU touches the data; both global sides are fully coalesced.

#define NB   32      // batch
#define HG   128     // H / P
#define WG   128     // W / P
#define CC   64      // channels
#define DIM  256     // P*P*C
#define JT   32      // j-tile (patches per workgroup)
#define SEGS 128     // CC * P  (c, pi) output-row segments per tile
#define LSTRIDE 68   // 64 w_locals + 4 pad floats; 272B row keeps 16B alignment

__global__ __launch_bounds__(256)
void depatch_embed_kernel(const float* __restrict__ x, float* __restrict__ out) {
    __shared__ float lds[SEGS * LSTRIDE];   // 34,816 bytes

    const int tid = threadIdx.x;
    const int wg  = blockIdx.x;
    const int jb  = wg & 3;            // j-block: 0..3
    const int i   = (wg >> 2) & 127;   // patch row: 0..127
    const int n   = wg >> 9;           // batch: 0..31

    const size_t in_base  = (size_t)n * (WG * HG) * DIM;
    const size_t out_base = (size_t)n * CC * 256 * 256;

#if defined(__gfx1250__)
    // ---- load phase: global -> LDS (output layout), async DMA, 8B granularity
    // e = m*256+tid in [0,4096): jl = patch-in-tile (0..31), chunk = d/2 (0..127)
    // 32 lanes of a wave read 32 consecutive 8B chunks = 256B contiguous global.
#pragma unroll
    for (int m = 0; m < 16; ++m) {
        const int e     = m * 256 + tid;
        const int jl    = e >> 7;        // 0..31
        const int chunk = e & 127;       // seg index; d = 2*chunk
        const int patch = (jb * JT + jl) * HG + i;
        const unsigned long long ga =
            (unsigned long long)(x + in_base + (size_t)patch * DIM + 2 * chunk);
        const uint32_t la =
            (uint32_t)(uintptr_t)&lds[chunk * LSTRIDE + jl * 2];
        asm volatile("global_load_async_to_lds_b64 %0, %1, off"
                     :: "v"(la), "v"(ga) : "memory");
    }
    asm volatile("s_wait_asynccnt 0x0" ::: "memory");
    __syncthreads();

    // ---- store phase: LDS -> global, async DMA, 16B granularity
    // e = m*256+tid in [0,2048): seg = 2c+pi (0..127), q = 16B slot (0..15)
    // 16 lanes stream one 256B contiguous output row segment.
#pragma unroll
    for (int m = 0; m < 8; ++m) {
        const int e   = m * 256 + tid;
        const int seg = e >> 4;          // 0..127
        const int q   = e & 15;
        const int c   = seg >> 1;
        const int pi  = seg & 1;
        const unsigned long long ga = (unsigned long long)(
            out + out_base + ((size_t)c * 256 + (2 * i + pi)) * 256
                + jb * 64 + 4 * q);
        const uint32_t la =
            (uint32_t)(uintptr_t)&lds[seg * LSTRIDE + 4 * q];
        asm volatile("global_store_async_from_lds_b128 %0, %1, off"
                     :: "v"(ga), "v"(la) : "memory");
    }
    // S_ENDPGM performs an implicit S_WAIT_IDLE (all counters, incl. ASYNCcnt).
#else
    // Fallback (host pass / non-gfx1250): same tiling via VGPRs.
#pragma unroll 4
    for (int m = 0; m < 16; ++m) {
        const int e     = m * 256 + tid;
        const int jl    = e >> 7;
        const int chunk = e & 127;
        const int patch = (jb * JT + jl) * HG + i;
        const float2 v = *(const float2*)(x + in_base + (size_t)patch * DIM + 2 * chunk);
        *(float2*)&lds[chunk * LSTRIDE + jl * 2] = v;
    }
    __syncthreads();
#pragma unroll 4
    for (int m = 0; m < 8; ++m) {
        const int e   = m * 256 + tid;
        const int seg = e >> 4;
        const int q   = e & 15;
        const int c   = seg >> 1;
        const int pi  = seg & 1;
        const float4 v = *(const float4*)&lds[seg * LSTRIDE + 4 * q];
        *(float4*)(out + out_base + ((size_t)c * 256 + (2 * i + pi)) * 256
                   + jb * 64 + 4 * q) = v;
    }
#endif
}

extern "C" void kernel_launch(void* const* d_in, const int* in_sizes, int n_in,
                              void* d_out, int out_size, void* d_ws, size_t ws_size,
                              hipStream_t stream) {
    (void)in_sizes; (void)n_in; (void)d_ws; (void)ws_size; (void)out_size;
    const float* x   = (const float*)d_in[0];   // [32, 16384, 256] f32
    float*       out = (float*)d_out;           // [32, 64, 256, 256] f32
    // One workgroup per (n, i, j-block): 32 * 128 * 4 = 16384 blocks, 256 threads.
    const int blocks = NB * HG * (WG / JT);
    depatch_embed_kernel<<<blocks, 256, 0, stream>>>(x, out);
}